// CopyNetDecoderWithAttention_17042430231328
// MI455X (gfx1250) — compile-verified
//
#include <hip/hip_runtime.h>
#include <hip/hip_bf16.h>

#define V_   32000
#define X_   100
#define VX_  32100
#define H_   512
#define H2_  1024
#define E_   256
#define BS_  32
#define S_   256
#define T_   64
#define G3_  1536   // 3*H
#define KX_  1280   // E + 2H

typedef __attribute__((ext_vector_type(2))) float v2f;
typedef __attribute__((ext_vector_type(8))) float v8f;

// ---------------------------------------------------------------------------
// CDNA5 async global->LDS copy (ASYNCcnt-tracked, no VGPR round trip).
// LDS destination address = low 32 bits of the flat shared pointer
// (LDS aperture keeps the byte offset in addr[31:0], ISA 10.2).
// ---------------------------------------------------------------------------
__device__ __forceinline__ unsigned lds_off(const void* p) {
    return (unsigned)(unsigned long long)p;
}
__device__ __forceinline__ void async_ld_b128(unsigned lds, const float* g) {
    asm volatile("global_load_async_to_lds_b128 %0, %1, off"
                 :: "v"(lds), "v"(g) : "memory");
}
__device__ __forceinline__ void wait_async0() {
#if __has_builtin(__builtin_amdgcn_s_wait_asynccnt)
    __builtin_amdgcn_s_wait_asynccnt(0);
#else
    asm volatile("s_wait_asynccnt 0x0" ::: "memory");
#endif
}

// ---------------------------------------------------------------------------
// Generic wave32 WMMA fp32 GEMM:  C[M,N] = act(A[M,K] @ W[N,K]^T + bias[N])
// Block = 256 threads (8 waves). Block tile = 32 rows x 128 cols.
// Wave w owns cols [n0+16w, n0+16w+16); computes two 16x16 C tiles (M=32).
// K staged via double-buffered async-to-LDS copies in chunks of 32:
//   issue chunk k+1 -> buf^1, WMMA on buf, s_wait_asynccnt 0 + barrier, swap.
// Row pitch 36 floats keeps every B128 LDS destination 16-byte aligned.
// Requires: M % 32 == 0 (grid.y = M/32), N % 128 == 0 (grid.x = N/128), K % 32 == 0.
// ACT: 0 = none, 1 = tanh, 2 = exp (fused softmax numerator).
// ---------------------------------------------------------------------------
template <int ACT>
__global__ __launch_bounds__(256) void gemm_nt_kernel(
    const float* __restrict__ A, const float* __restrict__ W,
    const float* __restrict__ bias, float* __restrict__ C,
    int M, int N, int K)
{
    __shared__ float As[2][32][36];    //  9216 B
    __shared__ float Bs[2][128][36];   // 36864 B  (Bs[n][k], K-contiguous)

    const int tid  = threadIdx.x;
    const int wave = tid >> 5;
    const int lane = tid & 31;
    const int ln16 = lane & 15;      // A row / B col within tile
    const int half = lane >> 4;      // K sub-pair select (K0,K1 vs K2,K3)
    const int m0   = blockIdx.y * 32;
    const int n0   = blockIdx.x * 128;
    const int nb   = wave * 16 + ln16;

    // Per-thread staging assignment (5 async B128 per thread per K-chunk).
    const int arow = tid >> 3;             // 0..31
    const int ak4  = (tid & 7) * 4;        // 0..28
    const int bn   = tid >> 1;             // 0..127
    const int bkh  = (tid & 1) * 16;       // 0 or 16
    const float* Ab = A + (size_t)(m0 + arow) * K + ak4;
    const float* Wb = W + (size_t)(n0 + bn) * K + bkh;

    auto stage = [&](int buf, int k0) {
        async_ld_b128(lds_off(&As[buf][arow][ak4]), Ab + k0);
        #pragma unroll
        for (int j = 0; j < 16; j += 4)
            async_ld_b128(lds_off(&Bs[buf][bn][bkh + j]), Wb + k0 + j);
    };

    v8f c0 = {};
    v8f c1 = {};

    stage(0, 0);
    int cur = 0;
    for (int k0 = 0; k0 < K; k0 += 32) {
        wait_async0();          // current chunk's DMA landed in LDS
        __syncthreads();        // all waves see it; prev compute finished
        if (k0 + 32 < K) stage(cur ^ 1, k0 + 32);   // prefetch next chunk

        #pragma unroll
        for (int kk = 0; kk < 32; kk += 4) {
            const int kA = kk + 2 * half;
            v2f a0, a1, b;
            a0.x = As[cur][ln16][kA];        a0.y = As[cur][ln16][kA + 1];
            a1.x = As[cur][16 + ln16][kA];   a1.y = As[cur][16 + ln16][kA + 1];
            b.x  = Bs[cur][nb][kA];          b.y  = Bs[cur][nb][kA + 1];
            c0 = __builtin_amdgcn_wmma_f32_16x16x4_f32(false, a0, false, b,
                                                       (short)0, c0, false, false);
            c1 = __builtin_amdgcn_wmma_f32_16x16x4_f32(false, a1, false, b,
                                                       (short)0, c1, false, false);
        }
        cur ^= 1;
    }

    const int ng = n0 + wave * 16 + ln16;
    const float bv = bias[ng];
    #pragma unroll
    for (int i = 0; i < 8; ++i) {
        const int r0 = m0 + i + 8 * half;
        float v0 = c0[i] + bv;
        float v1 = c1[i] + bv;
        if (ACT == 1) { v0 = tanhf(v0);  v1 = tanhf(v1); }
        if (ACT == 2) { v0 = __expf(v0); v1 = __expf(v1); }
        C[(size_t)r0 * N + ng]        = v0;
        C[(size_t)(r0 + 16) * N + ng] = v1;
    }
}

// s0 = encoded[:, -1, :] @ W_bridge^T   (32 x 512, K = 1024)
__global__ void bridge_kernel(const float* __restrict__ encoded,
                              const float* __restrict__ W_bridge,
                              float* __restrict__ h)
{
    int i = blockIdx.x * blockDim.x + threadIdx.x;
    if (i >= BS_ * H_) return;
    int b = i / H_, j = i % H_;
    const float* e = encoded + ((size_t)b * S_ + (S_ - 1)) * H2_;
    const float* w = W_bridge + (size_t)j * H2_;
    float acc = 0.f;
    for (int k = 0; k < H2_; k += 4) {
        float4 ev = *(const float4*)(e + k);
        float4 wv = *(const float4*)(w + k);
        acc += ev.x * wv.x + ev.y * wv.y + ev.z * wv.z + ev.w * wv.w;
    }
    h[i] = acc;
}

// x = concat(E_trg[inputs[:, t]], ctx)   -> (32 x 1280)
__global__ void assemble_x_kernel(const int* __restrict__ inputs,
                                  const float* __restrict__ E_trg,
                                  const float* __restrict__ ctx,
                                  float* __restrict__ x, int t)
{
    int i = blockIdx.x * blockDim.x + threadIdx.x;
    if (i >= BS_ * KX_) return;
    int b = i / KX_, j = i % KX_;
    float v;
    if (j < E_) {
        int tok = inputs[b * T_ + t];
        v = E_trg[(size_t)tok * E_ + j];
    } else {
        v = ctx[b * H2_ + (j - E_)];
    }
    x[i] = v;
}

// GRU gate math: h <- (1-z)*n + z*h  (in-place, element-wise)
__global__ void gru_gate_kernel(const float* __restrict__ gi,
                                const float* __restrict__ gh,
                                float* __restrict__ h)
{
    int i = blockIdx.x * blockDim.x + threadIdx.x;
    if (i >= BS_ * H_) return;
    int b = i / H_, j = i % H_;
    const float* gib = gi + (size_t)b * G3_;
    const float* ghb = gh + (size_t)b * G3_;
    float r = 1.f / (1.f + expf(-(gib[j] + ghb[j])));
    float z = 1.f / (1.f + expf(-(gib[H_ + j] + ghb[H_ + j])));
    float n = tanhf(gib[2 * H_ + j] + r * ghb[2 * H_ + j]);
    h[i] = (1.f - z) * n + z * h[i];
}

// Per batch row: psi_c -> exp (copy-score numerators), psi_a -> softmax -> ctx.
__global__ __launch_bounds__(256) void attn_kernel(
    const float* __restrict__ encoded, const int* __restrict__ encoded_idx,
    const float* __restrict__ enc_scores, const float* __restrict__ enc_attn,
    const float* __restrict__ h, float* __restrict__ ec, float* __restrict__ ctx)
{
    int b = blockIdx.x;
    int tid = threadIdx.x;
    __shared__ float sh[H_];
    __shared__ float red[256];
    __shared__ float aw[S_];

    sh[tid]       = h[b * H_ + tid];
    sh[256 + tid] = h[b * H_ + 256 + tid];
    __syncthreads();

    const float* es = enc_scores + ((size_t)b * S_ + tid) * H_;
    const float* ea = enc_attn   + ((size_t)b * S_ + tid) * H_;
    float pc = 0.f, pa = 0.f;
    for (int k = 0; k < H_; k += 4) {
        float4 e1 = *(const float4*)(es + k);
        float4 e2 = *(const float4*)(ea + k);
        pc += e1.x * sh[k] + e1.y * sh[k + 1] + e1.z * sh[k + 2] + e1.w * sh[k + 3];
        pa += e2.x * sh[k] + e2.y * sh[k + 1] + e2.z * sh[k + 2] + e2.w * sh[k + 3];
    }
    float mask = (encoded_idx[b * S_ + tid] == 0) ? -1000.f : 0.f;
    pc += mask;
    pa += mask;
    ec[b * S_ + tid] = expf(pc);   // joint-softmax numerator (no max: logits ~O(1))

    // Attention softmax over S=256 (deterministic tree reductions).
    red[tid] = pa; __syncthreads();
    for (int off = 128; off > 0; off >>= 1) {
        if (tid < off) red[tid] = fmaxf(red[tid], red[tid + off]);
        __syncthreads();
    }
    float mx = red[0]; __syncthreads();
    float e = expf(pa - mx);
    red[tid] = e; __syncthreads();
    for (int off = 128; off > 0; off >>= 1) {
        if (tid < off) red[tid] += red[tid + off];
        __syncthreads();
    }
    aw[tid] = e / red[0];
    __syncthreads();

    // ctx[b] = a @ encoded[b]   (256 x 1024 matvec)
    for (int d = tid; d < H2_; d += 256) {
        const float* eb = encoded + (size_t)b * S_ * H2_ + d;
        float acc = 0.f;
        for (int s = 0; s < S_; ++s) acc += aw[s] * eb[(size_t)s * H2_];
        ctx[b * H2_ + d] = acc;
    }
}

// inv_denom[b] = 1 / (sum exp(psi_g) + sum exp(psi_c))  — deterministic reduce.
__global__ __launch_bounds__(256) void denom_kernel(
    const float* __restrict__ expg, const float* __restrict__ ec,
    float* __restrict__ inv_denom)
{
    int b = blockIdx.x;
    int tid = threadIdx.x;
    __shared__ float red[256];
    const float* p = expg + (size_t)b * V_;
    float acc = 0.f;
    for (int v = tid; v < V_; v += 256) acc += p[v];
    acc += ec[b * S_ + tid];
    red[tid] = acc; __syncthreads();
    for (int off = 128; off > 0; off >>= 1) {
        if (tid < off) red[tid] += red[tid + off];
        __syncthreads();
    }
    if (tid == 0) inv_denom[b] = 1.f / red[0];
}

// out[b, t, v] = prob_g (v < V) else 1e-4
__global__ void write_out_kernel(const float* __restrict__ expg,
                                 const float* __restrict__ inv_denom,
                                 float* __restrict__ out, int t)
{
    int i = blockIdx.x * blockDim.x + threadIdx.x;
    if (i >= BS_ * VX_) return;
    int b = i / VX_, v = i % VX_;
    float val = (v < V_) ? expg[(size_t)b * V_ + v] * inv_denom[b] : 1e-4f;
    out[((size_t)b * T_ + t) * VX_ + v] = val;
}

// Deterministic scatter-add of copy probabilities (one lane per batch row).
__global__ void scatter_kernel(const int* __restrict__ encoded_idx,
                               const float* __restrict__ ec,
                               const float* __restrict__ inv_denom,
                               float* __restrict__ out, int t)
{
    int b = threadIdx.x;
    if (b >= BS_) return;
    float inv = inv_denom[b];
    float* row = out + ((size_t)b * T_ + t) * VX_;
    for (int s = 0; s < S_; ++s) {
        int v = encoded_idx[b * S_ + s];
        row[v] += ec[b * S_ + s] * inv;
    }
}

extern "C" void kernel_launch(void* const* d_in, const int* in_sizes, int n_in,
                              void* d_out, int out_size, void* d_ws, size_t ws_size,
                              hipStream_t stream)
{
    const int*   inputs      = (const int*)  d_in[0];
    const float* encoded     = (const float*)d_in[1];
    const int*   encoded_idx = (const int*)  d_in[2];
    const float* W_bridge    = (const float*)d_in[3];
    const float* E_trg       = (const float*)d_in[4];
    const float* W_ih        = (const float*)d_in[5];
    const float* W_hh        = (const float*)d_in[6];
    const float* b_ih        = (const float*)d_in[7];
    const float* b_hh        = (const float*)d_in[8];
    const float* Wg_w        = (const float*)d_in[9];
    const float* Wg_b        = (const float*)d_in[10];
    const float* Wo_w        = (const float*)d_in[11];
    const float* Wo_b        = (const float*)d_in[12];
    const float* Wa_w        = (const float*)d_in[13];
    const float* Wa_b        = (const float*)d_in[14];
    float* out = (float*)d_out;

    // Workspace layout (fp32), ~38.5 MB total.
    float* ws = (float*)d_ws;
    float* enc_scores = ws;                         // 8192*512
    float* enc_attn   = enc_scores + 8192 * 512;    // 8192*512
    float* expg       = enc_attn   + 8192 * 512;    // 32*32000
    float* h          = expg + (size_t)BS_ * V_;    // 32*512
    float* ctx        = h    + BS_ * H_;            // 32*1024
    float* xbuf       = ctx  + BS_ * H2_;           // 32*1280
    float* gi         = xbuf + BS_ * KX_;           // 32*1536
    float* gh         = gi   + BS_ * G3_;           // 32*1536
    float* ec         = gh   + BS_ * G3_;           // 32*256
    float* inv_denom  = ec   + BS_ * S_;            // 32

    const dim3 blk(256);

    // One-time precompute: enc_scores / enc_attn (tanh GEMMs), bridge h0, ctx0=0.
    gemm_nt_kernel<1><<<dim3(H_ / 128, (BS_ * S_) / 32), blk, 0, stream>>>(
        encoded, Wo_w, Wo_b, enc_scores, BS_ * S_, H_, H2_);
    gemm_nt_kernel<1><<<dim3(H_ / 128, (BS_ * S_) / 32), blk, 0, stream>>>(
        encoded, Wa_w, Wa_b, enc_attn, BS_ * S_, H_, H2_);
    bridge_kernel<<<(BS_ * H_ + 255) / 256, blk, 0, stream>>>(encoded, W_bridge, h);
    hipMemsetAsync(ctx, 0, (size_t)BS_ * H2_ * sizeof(float), stream);

    // 64 sequential decode steps (all operands L2-resident after step 0).
    for (int t = 0; t < T_; ++t) {
        assemble_x_kernel<<<(BS_ * KX_ + 255) / 256, blk, 0, stream>>>(
            inputs, E_trg, ctx, xbuf, t);
        gemm_nt_kernel<0><<<dim3(G3_ / 128, 1), blk, 0, stream>>>(
            xbuf, W_ih, b_ih, gi, BS_, G3_, KX_);
        gemm_nt_kernel<0><<<dim3(G3_ / 128, 1), blk, 0, stream>>>(
            h, W_hh, b_hh, gh, BS_, G3_, H_);
        gru_gate_kernel<<<(BS_ * H_ + 255) / 256, blk, 0, stream>>>(gi, gh, h);
        gemm_nt_kernel<2><<<dim3(V_ / 128, 1), blk, 0, stream>>>(
            h, Wg_w, Wg_b, expg, BS_, V_, H_);
        attn_kernel<<<BS_, blk, 0, stream>>>(
            encoded, encoded_idx, enc_scores, enc_attn, h, ec, ctx);
        denom_kernel<<<BS_, blk, 0, stream>>>(expg, ec, inv_denom);
        write_out_kernel<<<(BS_ * VX_ + 255) / 256, blk, 0, stream>>>(
            expg, inv_denom, out, t);
        scatter_kernel<<<1, 32, 0, stream>>>(encoded_idx, ec, inv_denom, out, t);
    }
}